// TemporalLSTM_88321707475181
// MI455X (gfx1250) — compile-verified
//
#include <hip/hip_runtime.h>

// Problem constants (match the reference).
#define BATCH 32
#define TSEQ  512
#define INDIM 2048
#define HD    512
#define G4H   (4 * HD)       // 2048
#define MROWS (BATCH * TSEQ) // 16384
#define NBLK  32             // persistent scan blocks

// Pair-interleaved B layout: row stride per K-pair, in floats.
// 160 DWORDs/row -> stride mod 64 banks = 32, so half-waves reading kp and
// kp+1 hit disjoint bank halves: conflict-free ds_load_b64 B fragments.
#define BPITCH 160
// A-tile pad (floats/row) in the GEMM: 36 -> conflict-free b64 fragment
// reads (4*(9m)+kb distinct mod 64) and 16B-aligned rows for b128 staging.
#define APAD 36

typedef float v2f __attribute__((ext_vector_type(2)));
typedef float v8f __attribute__((ext_vector_type(8)));

// ---------------------------------------------------------------------------
// Fast activations via v_exp_f32 (exp2).
// ---------------------------------------------------------------------------
__device__ __forceinline__ float sigmoid_f(float x) {
  return 1.0f / (1.0f + exp2f(x * -1.44269504f));
}
__device__ __forceinline__ float tanh_f(float x) {
  x = fminf(fmaxf(x, -15.0f), 15.0f);
  float e = exp2f(x * 2.88539008f); // exp(2x)
  return (e - 1.0f) / (e + 1.0f);
}

// ---------------------------------------------------------------------------
// Device-scope grid barrier (arrive + generation spin). All NBLK blocks are
// co-resident (1 block / WGP due to LDS footprint), so spinning is safe.
// ---------------------------------------------------------------------------
__device__ __forceinline__ void grid_sync(unsigned* cnt, unsigned* gen, unsigned nblk) {
  __syncthreads();
  if (threadIdx.x == 0) {
    __threadfence();
    unsigned g0 = __hip_atomic_load(gen, __ATOMIC_ACQUIRE, __HIP_MEMORY_SCOPE_AGENT);
    if (__hip_atomic_fetch_add(cnt, 1u, __ATOMIC_ACQ_REL, __HIP_MEMORY_SCOPE_AGENT) == nblk - 1u) {
      __hip_atomic_store(cnt, 0u, __ATOMIC_RELAXED, __HIP_MEMORY_SCOPE_AGENT);
      __hip_atomic_store(gen, g0 + 1u, __ATOMIC_RELEASE, __HIP_MEMORY_SCOPE_AGENT);
    } else {
      while (__hip_atomic_load(gen, __ATOMIC_ACQUIRE, __HIP_MEMORY_SCOPE_AGENT) == g0) {
        __builtin_amdgcn_s_sleep(2);
      }
    }
  }
  __syncthreads();
}

// ---------------------------------------------------------------------------
// C[M,N] = A[M,K] * W[N,K]^T + bias1[N] + bias2[N]
// Block tile 64x64, 256 threads = 8 wave32s, each wave owns two 16x16 C tiles.
// K staged through LDS in chunks of 32 via b128 global loads; fp32 WMMA
// 16x16x4. A: [m][k] pad-36 rows. B: pair-interleaved [kp][n] float2 so every
// B fragment is a single aligned ds_load_b64 (no v_mov glue).
// ---------------------------------------------------------------------------
__global__ __launch_bounds__(256) void wmma_gemm_bias(
    const float* __restrict__ A, const float* __restrict__ W,
    const float* __restrict__ bias1, const float* __restrict__ bias2,
    float* __restrict__ C, int N, int K) {
  __shared__ __align__(16) float As[64 * APAD];      // [m][k], pad 36
  __shared__ __align__(16) float Bsf[16 * BPITCH];   // [kp][n] pair-interleaved

  const int tid  = threadIdx.x;
  const int lane = tid & 31;
  const int w    = tid >> 5;
  const int grp  = lane >> 4;   // half-wave: selects K pair / M+8
  const int l16  = lane & 15;

  const int bm = blockIdx.y * 64;
  const int bn = blockIdx.x * 64;

  const int mtile  = w & 3;          // 0..3 -> 64 rows
  const int ntile0 = (w >> 2) * 2;   // {0,2} -> two adjacent 16-col tiles

  const int m  = mtile * 16 + l16;
  const int n0 = ntile0 * 16 + l16;
  const int n1 = n0 + 16;

  v8f acc0 = {};
  v8f acc1 = {};

  for (int k0 = 0; k0 < K; k0 += 32) {
#pragma unroll
    for (int i = 0; i < 2; ++i) {            // A chunk: 64x32, b128 staging
      int idx = i * 256 + tid;
      int am = idx >> 3, ak = (idx & 7) * 4;
      float4 av = *(const float4*)&A[(size_t)(bm + am) * K + k0 + ak];
      *(float4*)&As[am * APAD + ak] = av;
    }
#pragma unroll
    for (int i = 0; i < 2; ++i) {            // B chunk: b128 load -> 2x b64 LDS
      int idx = i * 256 + tid;
      int wn = idx >> 3, wk = (idx & 7) * 4;
      float4 wv = *(const float4*)&W[(size_t)(bn + wn) * K + k0 + wk];
      int kp = wk >> 1;
      v2f lo; lo.x = wv.x; lo.y = wv.y;
      v2f hi; hi.x = wv.z; hi.y = wv.w;
      *(v2f*)&Bsf[kp * BPITCH + wn * 2]       = lo;
      *(v2f*)&Bsf[(kp + 1) * BPITCH + wn * 2] = hi;
    }
    __syncthreads();

#pragma unroll
    for (int ks = 0; ks < 8; ++ks) {
      int kb = ks * 4 + grp * 2;   // A k-offset
      int kp = ks * 2 + grp;       // B pair-row
      v2f a  = *(const v2f*)&As[m * APAD + kb];
      v2f b0 = *(const v2f*)&Bsf[kp * BPITCH + n0 * 2];
      v2f b1 = *(const v2f*)&Bsf[kp * BPITCH + n1 * 2];
      acc0 = __builtin_amdgcn_wmma_f32_16x16x4_f32(false, a, false, b0, (short)0, acc0, false, false);
      acc1 = __builtin_amdgcn_wmma_f32_16x16x4_f32(false, a, false, b1, (short)0, acc1, false, false);
    }
    __syncthreads();
  }

  float bv0 = bias1[bn + n0] + bias2[bn + n0];
  float bv1 = bias1[bn + n1] + bias2[bn + n1];
#pragma unroll
  for (int r = 0; r < 8; ++r) {
    int row = bm + mtile * 16 + r + grp * 8;  // C layout: M = r + 8*half
    C[(size_t)row * N + bn + n0] = acc0[r] + bv0;
    C[(size_t)row * N + bn + n1] = acc1[r] + bv1;
  }
}

// ---------------------------------------------------------------------------
// Persistent LSTM scan. 32 blocks; block `blk` owns hidden units
// [blk*16, blk*16+16) -> 64 gate columns. W_hh slice lives pair-interleaved
// in LDS (160 KB) for the whole scan — only possible with CDNA5's 320 KB/WGP.
// Per step: stage h to LDS, 8 waves each compute one 16x16 gate tile over
// K=512 with fp32 WMMA (dual accumulators), activations, cross-wave combine
// via LDS, cell update, publish h, grid barrier.
// xg rows / out rows are indexed (b*T + t).
// ---------------------------------------------------------------------------
__global__ __launch_bounds__(256) void lstm_scan(
    const float* __restrict__ xg,    // [B*T, 4H]
    const float* __restrict__ Whh,   // [4H, H]
    float* __restrict__ hbuf,        // [B, H] global h ping buffer
    float* __restrict__ out,         // [B*T, H]
    unsigned* cnt, unsigned* gen) {
  __shared__ __align__(16) float Bsf[(HD / 2) * BPITCH]; // W_hh pairs, 160 KB
  __shared__ __align__(16) float As[BATCH * 516];        // staged h [b][k]
  __shared__ __align__(16) float GS[4 * BATCH * 16];     // activated gates
  __shared__ __align__(16) float CS[BATCH * 16];         // cell state slice

  const int tid  = threadIdx.x;
  const int lane = tid & 31;
  const int w    = tid >> 5;
  const int grp  = lane >> 4;
  const int l16  = lane & 15;
  const int blk  = blockIdx.x;
  const int j0   = blk * 16;

  const int g     = w >> 1;             // gate 0..3 (i,f,g,o)
  const int mtile = w & 1;              // batch half
  const int m     = mtile * 16 + l16;   // batch row for A fragment
  const int nb2   = (g * 16 + l16) * 2; // float index of this lane's B pair

  // Init cell state + own h slice.
  for (int e = tid; e < BATCH * 16; e += 256) {
    int b = e >> 4, j = e & 15;
    CS[e] = 0.0f;
    hbuf[b * HD + j0 + j] = 0.0f;
  }
  // Load W_hh slice into pair-interleaved LDS:
  // local col rl = g*16+j  <->  Whh row g*H + j0 + j; pair row kp = k>>1.
  for (int e = tid; e < 64 * HD; e += 256) {
    int rl = e >> 9;
    int k  = e & (HD - 1);
    int gg = rl >> 4, j = rl & 15;
    Bsf[(k >> 1) * BPITCH + rl * 2 + (k & 1)] =
        Whh[(size_t)(gg * HD + j0 + j) * HD + k];
  }
  __threadfence();
  grid_sync(cnt, gen, NBLK);   // all h zeros + weights visible

  for (int t = 0; t < TSEQ; ++t) {
    // Prefetch this step's xg gate values; they are consumed right after the
    // WMMA loop, so the prefetch hides HBM latency behind the matmul.
    const size_t xgbase = (size_t)t * G4H + g * HD + j0 + l16;
#pragma unroll
    for (int r = 0; r < 8; ++r) {
      int b = mtile * 16 + r + grp * 8;
      __builtin_prefetch(&xg[(size_t)b * TSEQ * G4H + xgbase], 0, 1);
    }

    // Stage full h [32,512] into LDS (coalesced b128 loads).
    for (int e = tid; e < BATCH * HD / 4; e += 256) {
      int flat = e * 4;
      int b = flat >> 9, k = flat & (HD - 1);
      *(float4*)&As[b * 516 + k] = *(const float4*)&hbuf[flat];
    }
    __syncthreads();

    // 16x16 gate tile over K=512; dual accumulators break the C chain.
    v8f accA = {};
    v8f accB = {};
#pragma unroll 4
    for (int ks = 0; ks < HD / 8; ++ks) {
      int kb0 = ks * 8 + grp * 2;   // A k-offsets
      int kb1 = kb0 + 4;
      int kp0 = ks * 4 + grp;       // B pair-rows
      int kp1 = kp0 + 2;
      v2f a0 = *(const v2f*)&As[m * 516 + kb0];
      v2f a1 = *(const v2f*)&As[m * 516 + kb1];
      v2f b0 = *(const v2f*)&Bsf[kp0 * BPITCH + nb2];
      v2f b1 = *(const v2f*)&Bsf[kp1 * BPITCH + nb2];
      accA = __builtin_amdgcn_wmma_f32_16x16x4_f32(false, a0, false, b0, (short)0, accA, false, false);
      accB = __builtin_amdgcn_wmma_f32_16x16x4_f32(false, a1, false, b1, (short)0, accB, false, false);
    }
    v8f acc = accA + accB;

    // Add precomputed input gates, activate, publish to LDS.
#pragma unroll
    for (int r = 0; r < 8; ++r) {
      int b = mtile * 16 + r + grp * 8;
      float v = acc[r] + xg[(size_t)b * TSEQ * G4H + xgbase];
      v = (g == 2) ? tanh_f(v) : sigmoid_f(v);
      GS[(g * BATCH + b) * 16 + l16] = v;
    }
    __syncthreads();

    // Cell/hidden update for the owned 32x16 slice.
    for (int e = tid; e < BATCH * 16; e += 256) {
      int b = e >> 4, j = e & 15;
      float iv = GS[(0 * BATCH + b) * 16 + j];
      float fv = GS[(1 * BATCH + b) * 16 + j];
      float gv = GS[(2 * BATCH + b) * 16 + j];
      float ov = GS[(3 * BATCH + b) * 16 + j];
      float c  = fv * CS[e] + iv * gv;
      CS[e] = c;
      float h = ov * tanh_f(c);
      hbuf[b * HD + j0 + j] = h;
      out[((size_t)b * TSEQ + t) * HD + j0 + j] = h;
    }
    __threadfence();
    grid_sync(cnt, gen, NBLK);
  }
}

// ---------------------------------------------------------------------------
// Launch: gemm(xg0) -> scan(layer0) -> gemm(xg1, reusing xg buffer) -> scan.
// Workspace: xg (134MB, reused) + h1 (34MB) + hbuf (64KB) + barrier.
// ---------------------------------------------------------------------------
extern "C" void kernel_launch(void* const* d_in, const int* in_sizes, int n_in,
                              void* d_out, int out_size, void* d_ws, size_t ws_size,
                              hipStream_t stream) {
  (void)in_sizes; (void)n_in; (void)out_size; (void)ws_size;
  const float* X     = (const float*)d_in[0];
  const float* W_ih0 = (const float*)d_in[1];
  const float* W_hh0 = (const float*)d_in[2];
  const float* b_ih0 = (const float*)d_in[3];
  const float* b_hh0 = (const float*)d_in[4];
  const float* W_ih1 = (const float*)d_in[5];
  const float* W_hh1 = (const float*)d_in[6];
  const float* b_ih1 = (const float*)d_in[7];
  const float* b_hh1 = (const float*)d_in[8];
  float* out = (float*)d_out;

  float* xg   = (float*)d_ws;                       // [16384, 2048]
  float* h1   = xg + (size_t)MROWS * G4H;           // [16384, 512]
  float* hbuf = h1 + (size_t)MROWS * HD;            // [32, 512]
  unsigned* bar = (unsigned*)(hbuf + BATCH * HD);   // {cnt, gen}

  hipMemsetAsync(bar, 0, 2 * sizeof(unsigned), stream);

  dim3 gg(G4H / 64, MROWS / 64);  // (32, 256)
  wmma_gemm_bias<<<gg, 256, 0, stream>>>(X, W_ih0, b_ih0, b_hh0, xg, G4H, INDIM);
  lstm_scan<<<NBLK, 256, 0, stream>>>(xg, W_hh0, hbuf, h1, bar, bar + 1);
  wmma_gemm_bias<<<gg, 256, 0, stream>>>(h1, W_ih1, b_ih1, b_hh1, xg, G4H, HD);
  lstm_scan<<<NBLK, 256, 0, stream>>>(xg, W_hh1, hbuf, out, bar, bar + 1);
}